// DYAN_81690277970354
// MI455X (gfx1250) — compile-verified
//
#include <hip/hip_runtime.h>
#include <hip/hip_bf16.h>

typedef __attribute__((ext_vector_type(2))) float v2f;
typedef __attribute__((ext_vector_type(8))) float v8f;

#define NPOLE 80
#define NP    161          // 2N+1
#define NPP   176          // 11*16
#define TSEQ  36
#define TSEQP 48
#define DDAT  50
#define DDATP 64           // 4*16
#define LAMF  0.1f
#define RMINC 0.8f
#define RMAXC 1.1f
#define TMINC 0.1f
#define TMAXC 3.14159265358979323846f
#define MAXIT 100

#define SY    72           // row stride (floats) for y/C arrays: 2*SY % 64 == 16 -> half-waves hit disjoint banks
#define SYIN  72
#define NTHR  512
#define NWAV  16

// LDS layout (float offsets)
#define OFF_A    0                        // [176*176]  DtD then A (symmetric)
#define OFF_D    (OFF_A   + NPP*NPP)      // [48*176]   D, zero padded
#define OFF_YIN  (OFF_D   + TSEQP*NPP)    // [36*72]    Y[n], zero padded
#define OFF_Y    (OFF_YIN + TSEQ*SYIN)    // [176*72]   FISTA y (also power-iter scratch)
#define OFF_C    (OFF_Y   + NPP*SY)       // [176*72]   C / reweight scratch
#define OFF_SUM  (OFF_C   + NPP*SY)       // [64]
#define OFF_SCAL (OFF_SUM + 64)           // [4]
#define LDS_FLOATS (OFF_SCAL + 4)

__device__ __forceinline__ v8f wmma_f32(v2f a, v2f b, v8f c) {
  // V_WMMA_F32_16X16X4_F32 : D = A(16x4,f32) * B(4x16,f32) + C(16x16,f32)
  return __builtin_amdgcn_wmma_f32_16x16x4_f32(false, a, false, b, (short)0, c, false, false);
}

__global__ void __launch_bounds__(NTHR)
dyan_fista_kernel(const float* __restrict__ X, const float* __restrict__ rho,
                  const float* __restrict__ theta, float* __restrict__ out, int ns)
{
  extern __shared__ float sm[];
  float* sA   = sm + OFF_A;
  float* sD   = sm + OFF_D;
  float* sYin = sm + OFF_YIN;
  float* sY   = sm + OFF_Y;
  float* sC   = sm + OFF_C;
  float* sSum = sm + OFF_SUM;
  float* sScal= sm + OFF_SCAL;

  const int tid  = threadIdx.x;
  const int n    = blockIdx.x;
  const int wave = tid >> 5;
  const int lane = tid & 31;
  const int kh   = lane >> 4;    // which K half the lane holds
  const int nl   = lane & 15;

  // ---------------- build D (clamped poles), zero-padded ----------------
  for (int idx = tid; idx < TSEQP * NPP; idx += NTHR) {
    int t = idx / NPP, p = idx % NPP;
    float v = 0.f;
    if (t < TSEQ && p < NP) {
      if (p == 0) v = 1.f;
      else {
        int j = (p <= NPOLE) ? (p - 1) : (p - 1 - NPOLE);
        float r  = fminf(fmaxf(rho[j],   RMINC), RMAXC);
        float th = fminf(fmaxf(theta[j], TMINC), TMAXC);
        float pw = __expf((float)t * __logf(r));        // r^t, r in [0.8,1.1]
        float ang = th * (float)t;
        v = pw * ((p <= NPOLE) ? cosf(ang) : sinf(ang));
      }
    }
    sD[idx] = v;
  }
  // load Y[n] into LDS, zero-padded columns
  for (int idx = tid; idx < TSEQ * SYIN; idx += NTHR) {
    int t = idx / SYIN, d = idx % SYIN;
    sYin[idx] = (d < DDAT) ? X[(size_t)n * TSEQ * DDAT + t * DDAT + d] : 0.f;
  }
  __syncthreads();

  // ---------------- DtD -> sA (padded region stays 0) ----------------
  for (int idx = tid; idx < NPP * NPP; idx += NTHR) {
    int p = idx / NPP, q = idx % NPP;
    float s = 0.f;
    for (int t = 0; t < TSEQ; ++t) s += sD[t * NPP + p] * sD[t * NPP + q];
    sA[idx] = s;
  }
  __syncthreads();

  // ---------------- L = ||DtD||_2 via power iteration ----------------
  float* pv = sY;        // v vector scratch
  float* pu = sY + 256;  // u vector scratch
  if (tid < NPP) pv[tid] = (tid < NP) ? 1.f : 0.f;
  __syncthreads();
  for (int it = 0; it < 96; ++it) {
    if (tid < NP) {
      float s = 0.f;
      for (int q = 0; q < NP; ++q) s += sA[tid * NPP + q] * pv[q];
      pu[tid] = s;
    }
    __syncthreads();
    if (tid == 0) {
      float s = 0.f;
      for (int q = 0; q < NP; ++q) s += pu[q] * pu[q];
      sScal[0] = sqrtf(s);
    }
    __syncthreads();
    float nrm = sScal[0];
    if (tid < NP) pv[tid] = pu[tid] / nrm;
    __syncthreads();
  }
  const float Linv = 1.f / sScal[0];
  __syncthreads();

  // ---------------- A = I - DtD*Linv (in place, pads -> 0), zero y ----------------
  for (int idx = tid; idx < NPP * NPP; idx += NTHR) {
    int p = idx / NPP, q = idx % NPP;
    float a = -sA[idx] * Linv;
    if (p == q && p < NP) a += 1.f;
    sA[idx] = a;
  }
  for (int idx = tid; idx < NPP * SY; idx += NTHR) sY[idx] = 0.f;
  __syncthreads();

  // ---------------- tile assignment: 44 tiles over 16 waves, <=3 per wave ----------------
  int  mcol[3], ncol[3], pbase[3];
  bool valid[3];
#pragma unroll
  for (int i = 0; i < 3; ++i) {
    int tix  = wave + NWAV * i;
    valid[i] = (tix < 44);
    int tc   = valid[i] ? tix : 0;
    int mT   = tc >> 2, nT = tc & 3;
    mcol[i]  = mT * 16 + nl;        // output row (via lane) / A-frag M
    ncol[i]  = nT * 16 + nl;        // output col d / B-frag N
    pbase[i] = mT * 16 + 8 * kh;    // C/D frag: VGPR r -> row pbase+r
  }

  // ---------------- DtY*Linv via WMMA (A-operand = D^T, read transposed) ----------------
  v8f z = {0.f,0.f,0.f,0.f,0.f,0.f,0.f,0.f};
  v8f bacc[3] = {z, z, z};
#pragma unroll 1
  for (int kk = 0; kk < 9; ++kk) {          // K = 36 = 9*4
    int kr = kk * 4 + 2 * kh;
#pragma unroll
    for (int i = 0; i < 3; ++i) {
      v2f a, b;
      a.x = sD[(kr + 0) * NPP + mcol[i]];   // D^T[p][t] = D[t][p]
      a.y = sD[(kr + 1) * NPP + mcol[i]];
      b.x = sYin[(kr + 0) * SYIN + ncol[i]];
      b.y = sYin[(kr + 1) * SYIN + ncol[i]];
      bacc[i] = wmma_f32(a, b, bacc[i]);
    }
  }

  float breg[3][8], xreg[3][8], wlreg[3][8];
#pragma unroll
  for (int i = 0; i < 3; ++i)
#pragma unroll
    for (int r = 0; r < 8; ++r) {
      breg[i][r]  = bacc[i][r] * Linv;   // DtY/L
      xreg[i][r]  = 0.f;
      wlreg[i][r] = LAMF * Linv;         // w=1 initially
    }

  // ---------------- 2 reweighting rounds x 100 FISTA iterations ----------------
  for (int round = 0; round < 2; ++round) {
    float tf = 1.f;
#pragma unroll 1
    for (int it = 0; it < MAXIT; ++it) {
      v8f acc[3] = {z, z, z};
#pragma unroll 1
      for (int kk = 0; kk < 44; ++kk) {   // K = 176 = 44*4
        int kr = kk * 4 + 2 * kh;
#pragma unroll
        for (int i = 0; i < 3; ++i) {
          v2f a, b;
          a.x = sA[(kr + 0) * NPP + mcol[i]];   // A symmetric -> transposed read
          a.y = sA[(kr + 1) * NPP + mcol[i]];
          b.x = sY[(kr + 0) * SY + ncol[i]];
          b.y = sY[(kr + 1) * SY + ncol[i]];
          acc[i] = wmma_f32(a, b, acc[i]);
        }
      }
      __syncthreads();                     // everyone done reading y
      float tn  = 0.5f * (1.f + sqrtf(1.f + 4.f * tf * tf));
      float ttc = (tf - 1.f) / tn;
      tf = tn;
#pragma unroll
      for (int i = 0; i < 3; ++i) {
        if (valid[i]) {
#pragma unroll
          for (int r = 0; r < 8; ++r) {
            float az = acc[i][r] + breg[i][r];
            float xn = fmaxf(az - wlreg[i][r], 0.f) + fminf(az + wlreg[i][r], 0.f);
            float yn = xn + ttc * (xn - xreg[i][r]);
            xreg[i][r] = xn;
            sY[(pbase[i] + r) * SY + ncol[i]] = yn;
          }
        }
      }
      __syncthreads();                     // y_new visible for next iteration
    }

    if (round == 0) {
      // ---- deterministic reweight: w = (1/(|C|+.01)) / sum_p * Np ----
#pragma unroll
      for (int i = 0; i < 3; ++i)
        if (valid[i])
#pragma unroll
          for (int r = 0; r < 8; ++r)
            sC[(pbase[i] + r) * SY + ncol[i]] = 1.f / (fabsf(xreg[i][r]) + 0.01f);
      __syncthreads();
      if (tid < DDATP) {
        float s = 0.f;
        for (int p = 0; p < NP; ++p) s += sC[p * SY + tid];
        sSum[tid] = s;
      }
      __syncthreads();
#pragma unroll
      for (int i = 0; i < 3; ++i)
#pragma unroll
        for (int r = 0; r < 8; ++r) {
          float wraw = 1.f / (fabsf(xreg[i][r]) + 0.01f);
          float w    = wraw / sSum[ncol[i]] * (float)NP;
          wlreg[i][r] = w * (LAMF * Linv);
          xreg[i][r]  = 0.f;               // x0 = 0 for round 2
        }
      for (int idx = tid; idx < NPP * SY; idx += NTHR) sY[idx] = 0.f;  // y0 = 0
      __syncthreads();
    }
  }

  // ---------------- C -> LDS, store C, store D (WG 0) ----------------
#pragma unroll
  for (int i = 0; i < 3; ++i)
    if (valid[i])
#pragma unroll
      for (int r = 0; r < 8; ++r)
        sC[(pbase[i] + r) * SY + ncol[i]] = xreg[i][r];
  __syncthreads();

  const size_t Cbase = 0;
  const size_t Dbase = (size_t)ns * NP * DDAT;
  const size_t Rbase = Dbase + (size_t)TSEQ * NP;

  for (int idx = tid; idx < NP * DDAT; idx += NTHR) {
    int p = idx / DDAT, d = idx % DDAT;
    out[Cbase + (size_t)n * NP * DDAT + idx] = sC[p * SY + d];
  }
  if (n == 0) {
    for (int idx = tid; idx < TSEQ * NP; idx += NTHR) {
      int t = idx / NP, p = idx % NP;
      out[Dbase + idx] = sD[t * NPP + p];
    }
  }

  // ---------------- R = D @ C via WMMA : 12 tiles (3 mtiles x 4 ntiles) ----------------
  {
    bool rv = (wave < 12);
    int  w2 = rv ? wave : 0;
    int  mT = w2 >> 2, nT = w2 & 3;
    int  mrow = mT * 16 + nl;          // A-frag row t
    int  ncl  = nT * 16 + nl;
    v8f racc = z;
#pragma unroll 1
    for (int kk = 0; kk < 44; ++kk) {
      int kr = kk * 4 + 2 * kh;
      v2f a, b;
      a.x = sD[mrow * NPP + kr + 0];   // natural row-major read of D
      a.y = sD[mrow * NPP + kr + 1];
      b.x = sC[(kr + 0) * SY + ncl];
      b.y = sC[(kr + 1) * SY + ncl];
      racc = wmma_f32(a, b, racc);
    }
    if (rv) {
#pragma unroll
      for (int r = 0; r < 8; ++r) {
        int t = mT * 16 + 8 * kh + r;
        int d = ncl;
        if (t < TSEQ && d < DDAT)
          out[Rbase + (size_t)n * TSEQ * DDAT + t * DDAT + d] = racc[r];
      }
    }
  }
}

extern "C" void kernel_launch(void* const* d_in, const int* in_sizes, int n_in,
                              void* d_out, int out_size, void* d_ws, size_t ws_size,
                              hipStream_t stream) {
  (void)n_in; (void)out_size; (void)d_ws; (void)ws_size;
  const float* X     = (const float*)d_in[0];
  const float* rho   = (const float*)d_in[1];
  const float* theta = (const float*)d_in[2];
  float* out = (float*)d_out;
  const int ns = in_sizes[0] / (TSEQ * DDAT);   // 128
  const size_t lds_bytes = (size_t)LDS_FLOATS * sizeof(float);  // ~264 KB, fits 320 KB/WGP
  dyan_fista_kernel<<<dim3(ns), dim3(NTHR), lds_bytes, stream>>>(X, rho, theta, out, ns);
}